// Model_6330781794321
// MI455X (gfx1250) — compile-verified
//
#include <hip/hip_runtime.h>
#include <cmath>

#define T_DIM 2048
#define B_DIM 256
#define F_DIM 128
#define H_DIM 50
#define H3    150
#define NP    160      // padded 3H (10 WMMA n-tiles)
#define NCLS  8
#define ALPHA 0.3f

typedef float v2f __attribute__((ext_vector_type(2)));
typedef float v8f __attribute__((ext_vector_type(8)));

// -------------------------------------------------------------------------
// Kernel 0: pad + K-pair-interleave W into workspace:
//   wpad[kk][col][j] = W[2*kk + j][col]  (col >= 150 -> 0), kk = 0..63
// This makes every GEMM B-operand a single unconditional global_load_b64.
// -------------------------------------------------------------------------
__global__ void __launch_bounds__(256)
pad_w(const float* __restrict__ W, float* __restrict__ wpad)
{
    int idx = blockIdx.x * 256 + threadIdx.x;        // 0 .. 64*160*2-1
    if (idx < 64 * NP * 2) {
        int kk  = idx / (NP * 2);
        int rem = idx % (NP * 2);
        int c = rem >> 1, j = rem & 1;
        wpad[idx] = (c < H3) ? W[(2 * kk + j) * H3 + c] : 0.0f;
    }
}

// -------------------------------------------------------------------------
// Kernel 1: xw[t*B + b, 0..159] = x[b,t,:] @ W + b0   (cols 150..159 -> 0)
// Block = 128 threads (4 waves); each wave: 16 rows x 160 cols (10 tiles).
// A tile (64x128 f32) staged in LDS (stride 132: 16B-aligned, bank-spread).
// B from wpad: 10 coalesced b64 loads per k-step, no branches, no EXEC ops.
// -------------------------------------------------------------------------
__global__ void __launch_bounds__(128)
gru_input_gemm(const float* __restrict__ x, const float* __restrict__ wpad,
               const float* __restrict__ bias0, float* __restrict__ xw)
{
    __shared__ float A[64 * 132];
    const int tid  = threadIdx.x;
    const int lane = tid & 31;
    const int wave = tid >> 5;
    const int ln   = lane & 15;
    const int hi   = lane >> 4;       // half-wave: 0 or 1
    const int koff = hi * 2;

    const int m0  = blockIdx.x * 64;  // row index m = t*256 + b ; 64 | 256
    const int t   = m0 >> 8;
    const int bb0 = m0 & 255;

    // cooperative coalesced load of A tile: 64 rows x 128 cols (float4)
    for (int i = 0; i < 16; ++i) {
        int idx = tid + i * 128;
        int r   = idx >> 5;
        int c4  = idx & 31;
        float4 v = *(const float4*)(x + ((size_t)(bb0 + r) * T_DIM + t) * F_DIM + c4 * 4);
        *(float4*)(&A[r * 132 + c4 * 4]) = v;
    }
    __syncthreads();

    // accumulators, bias-initialized (all rows of a tile share col bias)
    v8f acc[10];
#pragma unroll
    for (int nt = 0; nt < 10; ++nt) {
        int col  = nt * 16 + ln;
        float bv = (col < H3) ? bias0[col] : 0.0f;
        v8f a0 = {bv, bv, bv, bv, bv, bv, bv, bv};
        acc[nt] = a0;
    }

    const float* Arow  = &A[(wave * 16 + ln) * 132];
    // wpad offset: (k0/2 + hi)*2*NP + col*2  = wbase + k0*NP + nt*32
    const float* wbase = wpad + hi * (2 * NP) + ln * 2;

#pragma unroll 2
    for (int k0 = 0; k0 < F_DIM; k0 += 4) {
        v2f a = *(const v2f*)(Arow + k0 + koff);   // A: M=ln, K=k0+koff..+1
#pragma unroll
        for (int nt = 0; nt < 10; ++nt) {
            v2f b = *(const v2f*)(wbase + k0 * NP + nt * 32);
            acc[nt] = __builtin_amdgcn_wmma_f32_16x16x4_f32(
                false, a, false, b, (short)0, acc[nt], false, false);
        }
    }

    // store: lane owns column (n0+ln); vgpr v -> row (hi*8 + v)
    const int mrow = m0 + wave * 16 + hi * 8;
#pragma unroll
    for (int nt = 0; nt < 10; ++nt) {
        int col = nt * 16 + ln;
#pragma unroll
        for (int v = 0; v < 8; ++v)
            xw[(size_t)(mrow + v) * NP + col] = acc[nt][v];
    }
}

// -------------------------------------------------------------------------
// Kernel 2: sequential GRU scan (parallel over batch) + LeakyReLU+FC+softmax.
// Block = 320 threads (10 waves); wave w owns N-tile w of inner = h@U + b1.
// U tile lives in REGISTERS (26 floats/lane, loaded once); h in LDS.
// Per step: 13 x (ds_load_b64 + v_wmma) with a 2-way split accum chain.
// -------------------------------------------------------------------------
__global__ void __launch_bounds__(320)
gru_scan_head(const float* __restrict__ xw, const float* __restrict__ U,
              const float* __restrict__ bias1, const float* __restrict__ W2,
              const float* __restrict__ b2, float* __restrict__ out)
{
    __shared__ float Uli[26 * NP * 2];  // K-pair-interleaved, zero-padded
    __shared__ float hl[16 * 68];       // hidden state, cols 50..67 stay zero
    __shared__ float il[16 * NP];       // inner = h @ U + b1

    const int tid  = threadIdx.x;
    const int lane = tid & 31;
    const int ln   = lane & 15;
    const int hi   = lane >> 4;
    const int koff = hi * 2;
    const int n0   = (tid >> 5) * 16;   // wave's n-tile origin
    const int b0   = blockIdx.x * 16;

    for (int idx = tid; idx < 26 * NP * 2; idx += 320) {
        int kk  = idx / (NP * 2);
        int rem = idx % (NP * 2);
        int c = rem >> 1, j = rem & 1;
        int k = 2 * kk + j;
        Uli[idx] = (k < H_DIM && c < H3) ? U[k * H3 + c] : 0.0f;
    }
    for (int idx = tid; idx < 16 * 68; idx += 320) hl[idx] = 0.0f;
    __syncthreads();

    const int   col = n0 + ln;
    const float bv  = (col < H3) ? bias1[col] : 0.0f;

    // hoist the whole U tile for this wave into registers (once)
    v2f breg[13];
    {
        const float* ub = &Uli[hi * (NP * 2) + col * 2];
#pragma unroll
        for (int kk2 = 0; kk2 < 13; ++kk2)      // k0 = 4*kk2 -> pair 2*kk2+hi
            breg[kk2] = *(const v2f*)(ub + kk2 * (NP * 4));
    }

    // precompute per-thread gate/xw offsets (static across timesteps)
    int nel = 0, xoff[3], ioff[3], hoff[3];
    for (int e = tid; e < 16 * H_DIM; e += 320) {
        int r = e / H_DIM, c = e % H_DIM;
        xoff[nel] = (b0 + r) * NP + c;
        ioff[nel] = r * NP + c;
        hoff[nel] = r * 68 + c;
        ++nel;
    }

    const float* hrow  = &hl[ln * 68];
    size_t       tbase = 0;

    for (int t = 0; t < T_DIM; ++t) {
        // ---- fetch this step's x-projections (latency hides behind WMMA) --
        float xz[3], xr[3], xh[3];
#pragma unroll
        for (int i = 0; i < 3; ++i) {
            if (i < nel) {
                const float* p = xw + tbase + xoff[i];
                xz[i] = p[0];
                xr[i] = p[H_DIM];
                xh[i] = p[2 * H_DIM];
                if (t + 1 < T_DIM)                 // warm L2/WGP$ for t+1
                    __builtin_prefetch(p + (size_t)B_DIM * NP, 0, 3);
            }
        }

        // ---- inner tile: 13 WMMAs over K=52, 2-way split accum chain -----
        v8f acc0 = {bv, bv, bv, bv, bv, bv, bv, bv};
        v8f acc1 = {0.f, 0.f, 0.f, 0.f, 0.f, 0.f, 0.f, 0.f};
#pragma unroll
        for (int kk2 = 0; kk2 < 13; ++kk2) {
            v2f a = *(const v2f*)(hrow + 4 * kk2 + koff);
            if (kk2 & 1)
                acc1 = __builtin_amdgcn_wmma_f32_16x16x4_f32(
                    false, a, false, breg[kk2], (short)0, acc1, false, false);
            else
                acc0 = __builtin_amdgcn_wmma_f32_16x16x4_f32(
                    false, a, false, breg[kk2], (short)0, acc0, false, false);
        }
        v8f accs = acc0 + acc1;
#pragma unroll
        for (int v = 0; v < 8; ++v)
            il[(v + hi * 8) * NP + col] = accs[v];
        __syncthreads();

        // ---- gates ------------------------------------------------------
#pragma unroll
        for (int i = 0; i < 3; ++i) {
            if (i < nel) {
                float zi = 1.0f / (1.0f + __expf(-(xz[i] + il[ioff[i]])));
                float ri = 1.0f / (1.0f + __expf(-(xr[i] + il[ioff[i] + H_DIM])));
                float hh = tanhf(xh[i] + ri * il[ioff[i] + 2 * H_DIM]);
                float hp = hl[hoff[i]];
                hl[hoff[i]] = zi * hp + (1.0f - zi) * hh;
            }
        }
        __syncthreads();
        tbase += (size_t)B_DIM * NP;
    }

    // ---- head: LeakyReLU -> [16,50]x[50,8] -> softmax (1 thread / row) ---
    if (tid < 16) {
        float logits[NCLS];
#pragma unroll
        for (int j = 0; j < NCLS; ++j) logits[j] = b2[j];
        for (int c = 0; c < H_DIM; ++c) {
            float hv = hl[tid * 68 + c];
            hv = (hv >= 0.0f) ? hv : ALPHA * hv;
#pragma unroll
            for (int j = 0; j < NCLS; ++j) logits[j] += hv * W2[c * NCLS + j];
        }
        float mx = logits[0];
#pragma unroll
        for (int j = 1; j < NCLS; ++j) mx = fmaxf(mx, logits[j]);
        float ex[NCLS], s = 0.0f;
#pragma unroll
        for (int j = 0; j < NCLS; ++j) { ex[j] = __expf(logits[j] - mx); s += ex[j]; }
        float inv = 1.0f / s;
#pragma unroll
        for (int j = 0; j < NCLS; ++j)
            out[(size_t)(b0 + tid) * NCLS + j] = ex[j] * inv;
    }
}

// -------------------------------------------------------------------------
extern "C" void kernel_launch(void* const* d_in, const int* in_sizes, int n_in,
                              void* d_out, int out_size, void* d_ws, size_t ws_size,
                              hipStream_t stream) {
    const float* x  = (const float*)d_in[0];   // [256,2048,128]
    const float* W  = (const float*)d_in[1];   // [128,150]
    const float* U  = (const float*)d_in[2];   // [50,150]
    const float* b  = (const float*)d_in[3];   // [2,150]
    const float* W2 = (const float*)d_in[4];   // [50,8]
    const float* b2 = (const float*)d_in[5];   // [8]
    float* out  = (float*)d_out;               // [256,8]
    float* xw   = (float*)d_ws;                // [2048,256,160] = 320 MB
    float* wpad = xw + (size_t)T_DIM * B_DIM * NP;   // [64][160][2] = 80 KB

    pad_w<<<(64 * NP * 2 + 255) / 256, 256, 0, stream>>>(W, wpad);
    gru_input_gemm<<<(T_DIM * B_DIM) / 64, 128, 0, stream>>>(x, wpad, b, xw);
    gru_scan_head<<<B_DIM / 16, 320, 0, stream>>>(xw, U, b + H3, W2, b2, out);
}